// PoseVelGraph_10703058502092
// MI455X (gfx1250) — compile-verified
//
#include <hip/hip_runtime.h>
#include <hip/hip_bf16.h>
#include <cstdint>

#define EPSQ 1e-8f
#define TILE 256
#define TILE_FLOATS ((TILE + 1) * 7)   // 1799 floats staged per block

// ---------------------------------------------------------------------------
// Small float3/float4 helpers (scalar f32 math; workload is bandwidth-bound)
// ---------------------------------------------------------------------------
struct F3 { float x, y, z; };
struct F4 { float x, y, z, w; };

__device__ __forceinline__ F3 add3(F3 a, F3 b) { return {a.x + b.x, a.y + b.y, a.z + b.z}; }
__device__ __forceinline__ F3 sub3(F3 a, F3 b) { return {a.x - b.x, a.y - b.y, a.z - b.z}; }
__device__ __forceinline__ F3 scl3(F3 a, float s) { return {a.x * s, a.y * s, a.z * s}; }
__device__ __forceinline__ F3 neg3(F3 a) { return {-a.x, -a.y, -a.z}; }
__device__ __forceinline__ float dot3(F3 a, F3 b) { return a.x * b.x + a.y * b.y + a.z * b.z; }
__device__ __forceinline__ F3 cross3(F3 a, F3 b) {
  return {a.y * b.z - a.z * b.y, a.z * b.x - a.x * b.z, a.x * b.y - a.y * b.x};
}
__device__ __forceinline__ F4 qconj(F4 q) { return {-q.x, -q.y, -q.z, q.w}; }

// qmul(q, r): w = qw*rw - dot(qv,rv); v = qw*rv + rw*qv + cross(qv,rv)
__device__ __forceinline__ F4 qmul(F4 a, F4 b) {
  F3 av{a.x, a.y, a.z}, bv{b.x, b.y, b.z};
  F3 cr = cross3(av, bv);
  return {a.w * b.x + b.w * a.x + cr.x,
          a.w * b.y + b.w * a.y + cr.y,
          a.w * b.z + b.w * a.z + cr.z,
          a.w * b.w - dot3(av, bv)};
}

// qrot(q, v): t = 2*cross(qv, v); v + qw*t + cross(qv, t)
__device__ __forceinline__ F3 qrot(F4 q, F3 v) {
  F3 qv{q.x, q.y, q.z};
  F3 t = scl3(cross3(qv, v), 2.0f);
  F3 c2 = cross3(qv, t);
  return {v.x + q.w * t.x + c2.x, v.y + q.w * t.y + c2.y, v.z + q.w * t.z + c2.z};
}

__device__ __forceinline__ F3 so3_log(F4 q) {
  F3 v{q.x, q.y, q.z};
  float n = sqrtf(dot3(v, v));
  float theta = 2.0f * atan2f(n, q.w);
  float k;
  if (n > EPSQ) k = theta / n;
  else          k = 2.0f / ((fabsf(q.w) > EPSQ) ? q.w : 1.0f);
  return scl3(v, k);
}

// ---------------------------------------------------------------------------
// Edge kernel: per-edge SE(3) pose-graph residual (gather-heavy).
// nodes (28 MB) fits in the 192 MB L2 -> gathers use default RT policy.
// poses / output are streamed once -> non-temporal to avoid L2 pollution.
// ---------------------------------------------------------------------------
__global__ __launch_bounds__(256) void pg_edge_kernel(
    const int* __restrict__ edges, const float* __restrict__ nodes,
    const float* __restrict__ poses, float* __restrict__ out, int E) {
  int e = blockIdx.x * 256 + threadIdx.x;
  if (e >= E) return;

  const int2 idx = ((const int2*)edges)[e];
  const float* p1 = nodes + (size_t)idx.x * 7;
  const float* p2 = nodes + (size_t)idx.y * 7;
  const float* pp = poses + (size_t)e * 7;

  F3 t1{p1[0], p1[1], p1[2]};  F4 q1{p1[3], p1[4], p1[5], p1[6]};
  F3 t2{p2[0], p2[1], p2[2]};  F4 q2{p2[3], p2[4], p2[5], p2[6]};
  F3 tp{__builtin_nontemporal_load(pp + 0), __builtin_nontemporal_load(pp + 1),
        __builtin_nontemporal_load(pp + 2)};
  F4 qp{__builtin_nontemporal_load(pp + 3), __builtin_nontemporal_load(pp + 4),
        __builtin_nontemporal_load(pp + 5), __builtin_nontemporal_load(pp + 6)};

  F4 qi1 = qconj(q1);
  F3 ti1 = neg3(qrot(qi1, t1));
  F4 qa  = qmul(qi1, q2);
  F3 ta  = add3(ti1, qrot(qi1, t2));
  F4 qip = qconj(qp);
  F3 tip = neg3(qrot(qip, tp));
  F4 qe  = qmul(qip, qa);
  F3 te  = add3(tip, qrot(qip, ta));

  // se3_log(te, qe)
  F3 phi = so3_log(qe);
  float theta2 = dot3(phi, phi);
  float theta = sqrtf(theta2);
  float coef;
  if (theta < 1e-4f) coef = 1.0f / 12.0f;
  else coef = 1.0f / theta2 - (1.0f + cosf(theta)) / (2.0f * theta * sinf(theta));
  F3 pxt = cross3(phi, te);
  F3 ppx = cross3(phi, pxt);
  F3 tau{te.x - 0.5f * pxt.x + coef * ppx.x,
         te.y - 0.5f * pxt.y + coef * ppx.y,
         te.z - 0.5f * pxt.z + coef * ppx.z};

  float* o = out + (size_t)e * 6;   // L1 == 1.0
  __builtin_nontemporal_store(tau.x, o + 0);
  __builtin_nontemporal_store(tau.y, o + 1);
  __builtin_nontemporal_store(tau.z, o + 2);
  __builtin_nontemporal_store(phi.x, o + 3);
  __builtin_nontemporal_store(phi.y, o + 4);
  __builtin_nontemporal_store(phi.z, o + 5);
}

// ---------------------------------------------------------------------------
// IMU kernel: per-i residuals over consecutive node rows.
// Each block stages nodes[i0 .. i0+256] (1799 floats) into LDS via the CDNA5
// Tensor Data Mover (tensor_load_to_lds + s_wait_tensorcnt) when the builtin
// is available; tensor_dim0 is set to the remaining-element count so the TDM
// OOB rule (reads past tensor return zero) clamps the final tile for free.
// This toolchain (clang-23 / therock) exposes the 6-arg builtin:
//   (uint32x4 g0, int32x8 g1, int32x4 g2, int32x4 g3, int32x8 g4, i32 cpol)
// where the 5th group maps to the unused VADDR4 slot -> zero-filled.
// ---------------------------------------------------------------------------
#if defined(__AMDGCN__) && __has_builtin(__builtin_amdgcn_tensor_load_to_lds) && \
    __has_builtin(__builtin_amdgcn_s_wait_tensorcnt)
#define USE_TDM 1
typedef unsigned int v4u __attribute__((ext_vector_type(4)));
typedef int          v8i __attribute__((ext_vector_type(8)));
typedef int          v4i __attribute__((ext_vector_type(4)));
#else
#define USE_TDM 0
#endif

__global__ __launch_bounds__(256) void imu_kernel(
    const float* __restrict__ nodes, const float* __restrict__ vels,
    const float* __restrict__ imu_drots, const float* __restrict__ imu_dtrans,
    const float* __restrict__ imu_dvels, const float* __restrict__ dts,
    float* __restrict__ out1, float* __restrict__ out2, float* __restrict__ out3,
    int Mcnt, long long totalFloats) {
  __shared__ float lnodes[TILE_FLOATS];

  const int i0 = blockIdx.x * TILE;
  const long long baseFloat = (long long)i0 * 7;

#if USE_TDM
  if (threadIdx.x == 0) {
    unsigned long long ga = (unsigned long long)(uintptr_t)(nodes + baseFloat);
    unsigned ldsa = (unsigned)(uintptr_t)(void*)&lnodes[0];
    unsigned d0 = (unsigned)(totalFloats - baseFloat);  // remaining elements (OOB->0)
    // D# group 0: count=1 | lds_addr | global_addr[56:0] | type=2
    v4u g0 = {1u, ldsa, (unsigned)(ga & 0xFFFFFFFFu),
              (unsigned)((ga >> 32) & 0x1FFFFFFull) | 0x80000000u};
    // D# group 1: data_size=4B; tensor_dim0=d0; tensor_dim1=1; tile=1799x1;
    //             tensor_dim0_stride=d0
    v8i g1 = {(int)0x00020000u,
              (int)((d0 & 0xFFFFu) << 16),
              (int)(((d0 >> 16) & 0xFFFFu) | (1u << 16)),
              (int)((unsigned)TILE_FLOATS << 16),
              (int)1,
              (int)d0,
              0, 0};
    v4i g2 = {1, 1, 0, 0};                // tensor_dim2=1, tensor_dim3=1
    v4i g3 = {0, (int)(1u << 16), 0, 0};  // tensor_dim4=1
    v8i g4 = {0, 0, 0, 0, 0, 0, 0, 0};    // unused 5th SGPR group (VADDR4)
    __builtin_amdgcn_tensor_load_to_lds(g0, g1, g2, g3, g4, 0);
    __builtin_amdgcn_s_wait_tensorcnt(0);
  }
  __syncthreads();
#else
  for (int f = threadIdx.x; f < TILE_FLOATS; f += 256) {
    long long g = baseFloat + f;
    if (g < totalFloats) lnodes[f] = nodes[g];
  }
  __syncthreads();
#endif

  const int i = i0 + threadIdx.x;
  if (i >= Mcnt) return;

  const float* A = &lnodes[threadIdx.x * 7];
  const float* B = A + 7;
  F3 tA{A[0], A[1], A[2]};  F4 qA{A[3], A[4], A[5], A[6]};
  F3 tB{B[0], B[1], B[2]};  F4 qB{B[3], B[4], B[5], B[6]};

  const float* vp = vels + (size_t)i * 3;
  F3 vi {vp[0], vp[1], vp[2]};
  F3 vi1{vp[3], vp[4], vp[5]};

  // adjvelerr = imu_dvels - (vels[i+1] - vels[i]);  L2 = 0.1
  const float* dvp = imu_dvels + (size_t)i * 3;
  float* o1 = out1 + (size_t)i * 3;
  __builtin_nontemporal_store(0.1f * (__builtin_nontemporal_load(dvp + 0) - (vi1.x - vi.x)), o1 + 0);
  __builtin_nontemporal_store(0.1f * (__builtin_nontemporal_load(dvp + 1) - (vi1.y - vi.y)), o1 + 1);
  __builtin_nontemporal_store(0.1f * (__builtin_nontemporal_load(dvp + 2) - (vi1.z - vi.z)), o1 + 2);

  // imuroterr = so3_log( conj(imu_drots) * (conj(qA) * qB) );  L3 = 1.0
  const float* drp = imu_drots + (size_t)i * 4;
  F4 dr{__builtin_nontemporal_load(drp + 0), __builtin_nontemporal_load(drp + 1),
        __builtin_nontemporal_load(drp + 2), __builtin_nontemporal_load(drp + 3)};
  F4 qre = qmul(qconj(dr), qmul(qconj(qA), qB));
  F3 re = so3_log(qre);
  float* o2 = out2 + (size_t)i * 3;
  __builtin_nontemporal_store(re.x, o2 + 0);
  __builtin_nontemporal_store(re.y, o2 + 1);
  __builtin_nontemporal_store(re.z, o2 + 2);

  // transvelerr = tB - tA - (vels[i]*dt + imu_dtrans);  L4 = 0.1
  float dt = __builtin_nontemporal_load(dts + i);
  const float* dtp = imu_dtrans + (size_t)i * 3;
  F3 dtr{__builtin_nontemporal_load(dtp + 0), __builtin_nontemporal_load(dtp + 1),
         __builtin_nontemporal_load(dtp + 2)};
  float* o3 = out3 + (size_t)i * 3;
  __builtin_nontemporal_store(0.1f * (tB.x - tA.x - (vi.x * dt + dtr.x)), o3 + 0);
  __builtin_nontemporal_store(0.1f * (tB.y - tA.y - (vi.y * dt + dtr.y)), o3 + 1);
  __builtin_nontemporal_store(0.1f * (tB.z - tA.z - (vi.z * dt + dtr.z)), o3 + 2);
}

// ---------------------------------------------------------------------------
// Launch: out = [ L1*pgerr (6E) | L2*adjvelerr (3M) | L3*imuroterr (3M) |
//                 L4*transvelerr (3M) ]
// ---------------------------------------------------------------------------
extern "C" void kernel_launch(void* const* d_in, const int* in_sizes, int n_in,
                              void* d_out, int out_size, void* d_ws, size_t ws_size,
                              hipStream_t stream) {
  const int*   edges      = (const int*)d_in[0];
  const float* nodes      = (const float*)d_in[1];
  const float* vels       = (const float*)d_in[2];
  const float* poses      = (const float*)d_in[3];
  const float* imu_drots  = (const float*)d_in[4];
  const float* imu_dtrans = (const float*)d_in[5];
  const float* imu_dvels  = (const float*)d_in[6];
  const float* dts        = (const float*)d_in[7];

  const int E = in_sizes[0] / 2;
  const int N = in_sizes[1] / 7;
  const int M = N - 1;

  float* out  = (float*)d_out;
  float* out1 = out + (size_t)E * 6;
  float* out2 = out1 + (size_t)M * 3;
  float* out3 = out2 + (size_t)M * 3;

  pg_edge_kernel<<<(E + 255) / 256, 256, 0, stream>>>(edges, nodes, poses, out, E);
  imu_kernel<<<(M + TILE - 1) / TILE, 256, 0, stream>>>(
      nodes, vels, imu_drots, imu_dtrans, imu_dvels, dts,
      out1, out2, out3, M, (long long)N * 7);
}